// EntropyBottleneck_32736240730292
// MI455X (gfx1250) — compile-verified
//
#include <hip/hip_runtime.h>
#include <stdint.h>

#define NCH    192
#define PLANE  4096          // H*W = 64*64
#define NPLANE 1536          // B*C = 8*192
#define NTOT   (NPLANE * PLANE)

// ---------------- fast math helpers ----------------
__device__ __forceinline__ float fast_tanh(float x) {
#if __has_builtin(__builtin_amdgcn_tanhf)
  return __builtin_amdgcn_tanhf(x);              // gfx1250 v_tanh_f32
#elif __has_builtin(__builtin_amdgcn_tanh_f32)
  return __builtin_amdgcn_tanh_f32(x);
#else
  x = fminf(fmaxf(x, -15.0f), 15.0f);
  float e = __expf(2.0f * x);
  return (e - 1.0f) * __builtin_amdgcn_rcpf(e + 1.0f);
#endif
}

__device__ __forceinline__ float fast_sigmoid(float x) {
  return __builtin_amdgcn_rcpf(1.0f + __expf(-x));
}

__device__ __forceinline__ float softplus_f(float w) {
  // stable log1p(exp(w))
  float e = __expf(-fabsf(w));
  return fmaxf(w, 0.0f) + __logf(1.0f + e);
}

// ---------------- TDM (Tensor Data Mover) path ----------------
#if __has_builtin(__builtin_amdgcn_tensor_load_to_lds)
#define HAVE_TDM 1
typedef unsigned int v4u __attribute__((ext_vector_type(4)));
typedef int          v8i __attribute__((ext_vector_type(8)));
typedef int          v4i __attribute__((ext_vector_type(4)));

typedef __attribute__((address_space(3))) const void* lds_cptr_t;
__device__ __forceinline__ unsigned lds_byte_off(const void* p) {
  return (unsigned)(uintptr_t)(lds_cptr_t)p;
}

// 1-D tile: 4096 fp32 elements, contiguous, global -> LDS
__device__ __forceinline__ void tdm_load_plane(unsigned lds_off, const float* gsrc) {
  unsigned long long ga = (unsigned long long)(uintptr_t)gsrc;
  v4u g0;
  g0[0] = 1u;                                    // count=1, user descriptor
  g0[1] = lds_off;                               // lds_addr (bytes)
  g0[2] = (unsigned)(ga & 0xffffffffu);          // global_addr[31:0]
  g0[3] = (unsigned)((ga >> 32) & 0x1ffffffu)    // global_addr[56:32]
        | (2u << 30);                            // type = 2 ("image")
  v8i g1;
  g1[0] = (int)(2u << 16);                       // data_size = 4B, mask=0
  g1[1] = (int)(4096u << 16);                    // tensor_dim0[15:0] @ bits63:48
  g1[2] = (int)(1u << 16);                       // tensor_dim0 hi=0, tensor_dim1=1
  g1[3] = (int)(4096u << 16);                    // tile_dim0 = 4096
  g1[4] = 0;                                     // tile_dim1/2 unused
  g1[5] = 4096;                                  // tensor_dim0_stride (elements)
  g1[6] = 0;
  g1[7] = 0;
  v4i z4 = {0, 0, 0, 0};                         // groups 2/3 unused (<=2D tensor)
  v8i z8 = {0, 0, 0, 0, 0, 0, 0, 0};             // extra group (6-arg toolchain form)
  __builtin_amdgcn_tensor_load_to_lds(g0, g1, z4, z4, z8, 0);
}
#endif

// ---------------- per-channel MLP chain ----------------
// P layout: [0..2] SW0, [3..5] B0, [6..8] T0, [9..17] SW1, [18..20] B1,
// [21..23] T1, [24..32] SW2, [33..35] B2, [36..38] T2, [39..41] SW3, [42] B3
__device__ __forceinline__ float eval_chain(float v, const float* P) {
  float h0 = fmaf(P[0], v, P[3]); h0 = fmaf(P[6], fast_tanh(h0), h0);
  float h1 = fmaf(P[1], v, P[4]); h1 = fmaf(P[7], fast_tanh(h1), h1);
  float h2 = fmaf(P[2], v, P[5]); h2 = fmaf(P[8], fast_tanh(h2), h2);

  float g0 = P[18]; g0 = fmaf(P[9],  h0, g0); g0 = fmaf(P[10], h1, g0); g0 = fmaf(P[11], h2, g0);
  g0 = fmaf(P[21], fast_tanh(g0), g0);
  float g1 = P[19]; g1 = fmaf(P[12], h0, g1); g1 = fmaf(P[13], h1, g1); g1 = fmaf(P[14], h2, g1);
  g1 = fmaf(P[22], fast_tanh(g1), g1);
  float g2 = P[20]; g2 = fmaf(P[15], h0, g2); g2 = fmaf(P[16], h1, g2); g2 = fmaf(P[17], h2, g2);
  g2 = fmaf(P[23], fast_tanh(g2), g2);

  float k0 = P[33]; k0 = fmaf(P[24], g0, k0); k0 = fmaf(P[25], g1, k0); k0 = fmaf(P[26], g2, k0);
  k0 = fmaf(P[36], fast_tanh(k0), k0);
  float k1 = P[34]; k1 = fmaf(P[27], g0, k1); k1 = fmaf(P[28], g1, k1); k1 = fmaf(P[29], g2, k1);
  k1 = fmaf(P[37], fast_tanh(k1), k1);
  float k2 = P[35]; k2 = fmaf(P[30], g0, k2); k2 = fmaf(P[31], g1, k2); k2 = fmaf(P[32], g2, k2);
  k2 = fmaf(P[38], fast_tanh(k2), k2);

  float o = P[42];
  o = fmaf(P[39], k0, o); o = fmaf(P[40], k1, o); o = fmaf(P[41], k2, o);
  return o;
}

__device__ __forceinline__ float likelihood_of(float y, const float* P) {
  float lo = eval_chain(y - 0.5f, P);
  float up = eval_chain(y + 0.5f, P);
  float s  = lo + up;
  float sgn = (s > 0.0f) ? -1.0f : ((s < 0.0f) ? 1.0f : 0.0f);
  float l = fabsf(fast_sigmoid(sgn * up) - fast_sigmoid(sgn * lo));
  return fmaxf(l, 1e-9f);
}

// ---------------- main kernel: one block per (b,c) plane ----------------
__global__ __launch_bounds__(256) void eb_kernel(
    const float* __restrict__ x,  const float* __restrict__ noise,
    const float* __restrict__ w0, const float* __restrict__ b0, const float* __restrict__ f0,
    const float* __restrict__ w1, const float* __restrict__ b1, const float* __restrict__ f1,
    const float* __restrict__ w2, const float* __restrict__ b2, const float* __restrict__ f2,
    const float* __restrict__ w3, const float* __restrict__ b3,
    float* __restrict__ yout, float* __restrict__ lout)
{
  __shared__ __align__(16) float ldsx[PLANE];
  __shared__ __align__(16) float ldsn[PLANE];
  __shared__ float pbuf[48];

  const int plane = blockIdx.x;           // = b*NCH + c
  const int c     = plane % NCH;
  const int t     = threadIdx.x;
  const float* xg = x     + (size_t)plane * PLANE;
  const float* ng = noise + (size_t)plane * PLANE;

  // ---- precompute transformed per-channel parameters (softplus/tanh) ----
  if (t == 0) {
#pragma unroll
    for (int j = 0; j < 3; ++j) {
      pbuf[0  + j] = softplus_f(w0[c * 3 + j]);
      pbuf[3  + j] = b0[c * 3 + j];
      pbuf[6  + j] = fast_tanh(f0[c * 3 + j]);
      pbuf[18 + j] = b1[c * 3 + j];
      pbuf[21 + j] = fast_tanh(f1[c * 3 + j]);
      pbuf[33 + j] = b2[c * 3 + j];
      pbuf[36 + j] = fast_tanh(f2[c * 3 + j]);
      pbuf[39 + j] = softplus_f(w3[c * 3 + j]);
    }
#pragma unroll
    for (int j = 0; j < 9; ++j) {
      pbuf[9  + j] = softplus_f(w1[c * 9 + j]);
      pbuf[24 + j] = softplus_f(w2[c * 9 + j]);
    }
    pbuf[42] = b3[c];
  }

  // ---- stage x / noise tiles into LDS ----
#if defined(HAVE_TDM)
  if (t < 32) {                                 // wave 0 issues the DMAs
    tdm_load_plane(lds_byte_off(ldsx), xg);
    tdm_load_plane(lds_byte_off(ldsn), ng);
    __builtin_amdgcn_s_wait_tensorcnt(0);
  }
#else
  {
    const float4* xg4 = (const float4*)xg;
    const float4* ng4 = (const float4*)ng;
#pragma unroll
    for (int i = 0; i < 4; ++i) {
      int j = t + i * 256;
      ((float4*)ldsx)[j] = xg4[j];
      ((float4*)ldsn)[j] = ng4[j];
    }
  }
#endif
  __syncthreads();

  // ---- pull params into registers ----
  float P[43];
#pragma unroll
  for (int i = 0; i < 43; ++i) P[i] = pbuf[i];

  float4* yo = (float4*)(yout + (size_t)plane * PLANE);
  float4* lo = (float4*)(lout + (size_t)plane * PLANE);

#pragma unroll
  for (int i = 0; i < 4; ++i) {
    int j = t + i * 256;
    float4 xv = ((const float4*)ldsx)[j];
    float4 nv = ((const float4*)ldsn)[j];
    float4 yv, lv;
    yv.x = xv.x + nv.x; lv.x = likelihood_of(yv.x, P);
    yv.y = xv.y + nv.y; lv.y = likelihood_of(yv.y, P);
    yv.z = xv.z + nv.z; lv.z = likelihood_of(yv.z, P);
    yv.w = xv.w + nv.w; lv.w = likelihood_of(yv.w, P);
    yo[j] = yv;
    lo[j] = lv;
  }
}

extern "C" void kernel_launch(void* const* d_in, const int* in_sizes, int n_in,
                              void* d_out, int out_size, void* d_ws, size_t ws_size,
                              hipStream_t stream) {
  (void)in_sizes; (void)n_in; (void)d_ws; (void)ws_size; (void)out_size;
  const float* x     = (const float*)d_in[0];
  const float* noise = (const float*)d_in[1];
  const float* w0    = (const float*)d_in[2];
  const float* b0    = (const float*)d_in[3];
  const float* f0    = (const float*)d_in[4];
  const float* w1    = (const float*)d_in[5];
  const float* b1    = (const float*)d_in[6];
  const float* f1    = (const float*)d_in[7];
  const float* w2    = (const float*)d_in[8];
  const float* b2    = (const float*)d_in[9];
  const float* f2    = (const float*)d_in[10];
  const float* w3    = (const float*)d_in[11];
  const float* b3    = (const float*)d_in[12];
  float* yout = (float*)d_out;
  float* lout = (float*)d_out + (size_t)NTOT;

  eb_kernel<<<NPLANE, 256, 0, stream>>>(x, noise, w0, b0, f0, w1, b1, f1,
                                        w2, b2, f2, w3, b3, yout, lout);
}